// PerspectiveLayer_9208409882912
// MI455X (gfx1250) — compile-verified
//
#include <hip/hip_runtime.h>

// Perspective warp, bilinear, border=0.  B=64, H=W=512, C=3, f32.
// Memory-bound (~400 MB -> ~17us floor @ 23.3 TB/s). Strategy: per-workgroup
// output tile, async-stage the source bbox into LDS (CDNA5 ASYNCcnt path),
// gather taps from LDS (ds_load_b96). 1/det of inv(M) cancels in the
// projective divide, so the adjugate of M is used directly.

#define BI   64
#define HI   512
#define WI   512
#define CI   3
#define TW   64          // tile width  (pixels)
#define THT  32          // tile height (pixels)
#define LDSF 16384       // 64 KB of floats staged per workgroup

// ---------- CDNA5 async global->LDS staging (guarded) ----------------------
#if defined(__has_builtin)
# if __has_builtin(__builtin_amdgcn_global_load_async_to_lds_b128)
#  define HAVE_ASYNC_LDS 1
# endif
# if __has_builtin(__builtin_amdgcn_s_wait_asynccnt)
#  define HAVE_WAIT_ASYNC 1
# endif
#endif

typedef int v4i __attribute__((vector_size(4 * sizeof(int))));
typedef __attribute__((address_space(1))) v4i* gptr_v4i;   // global (prints as __device__)
typedef __attribute__((address_space(3))) v4i* lptr_v4i;   // LDS    (prints as __shared__)

__device__ __forceinline__ void g2l_b128(const float* g, float* l) {
#ifdef HAVE_ASYNC_LDS
  __builtin_amdgcn_global_load_async_to_lds_b128(
      (gptr_v4i)(g),
      (lptr_v4i)(l),
      /*offset=*/0, /*cpol=*/0);
#else
  l[0] = g[0]; l[1] = g[1]; l[2] = g[2]; l[3] = g[3];
#endif
}

__device__ __forceinline__ void async_wait_all() {
#ifdef HAVE_ASYNC_LDS
# ifdef HAVE_WAIT_ASYNC
  __builtin_amdgcn_s_wait_asynccnt(0);
# else
  asm volatile("s_wait_asynccnt 0" ::: "memory");
# endif
#endif
}

// ---------- bilinear tap (3 channels, border-constant 0) --------------------
// src points at the staged region origin (image coords ox,oy); reads are
// clamped into [cx0,cx1]x[cy0,cy1] (always inside the staged region); taps
// outside the image contribute zero via their weight.
__device__ __forceinline__ void bilinear3(const float* __restrict__ src, int pitch,
                                          int ox, int oy,
                                          int cx0, int cx1, int cy0, int cy1,
                                          float u, float v,
                                          float* __restrict__ dst) {
  float xf = floorf(u), yf = floorf(v);
  float wx = u - xf,  wy = v - yf;
  int x0 = (int)xf, y0 = (int)yf;
  int x1 = x0 + 1,  y1 = y0 + 1;
  float vx0 = ((unsigned)x0 < (unsigned)WI) ? 1.0f : 0.0f;
  float vx1 = ((unsigned)x1 < (unsigned)WI) ? 1.0f : 0.0f;
  float vy0 = ((unsigned)y0 < (unsigned)HI) ? 1.0f : 0.0f;
  float vy1 = ((unsigned)y1 < (unsigned)HI) ? 1.0f : 0.0f;
  float w00 = (1.0f - wy) * (1.0f - wx) * (vy0 * vx0);
  float w01 = (1.0f - wy) * wx          * (vy0 * vx1);
  float w10 = wy * (1.0f - wx)          * (vy1 * vx0);
  float w11 = wy * wx                   * (vy1 * vx1);
  int X0 = (min(max(x0, cx0), cx1) - ox) * CI;
  int X1 = (min(max(x1, cx0), cx1) - ox) * CI;
  const float* r0 = src + (min(max(y0, cy0), cy1) - oy) * pitch;
  const float* r1 = src + (min(max(y1, cy0), cy1) - oy) * pitch;
  float o0, o1, o2;   // 3 consecutive floats per tap -> b96 loads
  o0 = r0[X0 + 0] * w00 + r0[X1 + 0] * w01 + r1[X0 + 0] * w10 + r1[X1 + 0] * w11;
  o1 = r0[X0 + 1] * w00 + r0[X1 + 1] * w01 + r1[X0 + 1] * w10 + r1[X1 + 1] * w11;
  o2 = r0[X0 + 2] * w00 + r0[X1 + 2] * w01 + r1[X0 + 2] * w10 + r1[X1 + 2] * w11;
  dst[0] = o0; dst[1] = o1; dst[2] = o2;          // b96 store
}

__global__ __launch_bounds__(256)
void warp_persp_tiled(const float* __restrict__ imgs,
                      const float* __restrict__ M,
                      float* __restrict__ out) {
  __shared__ float lds[LDSF];

  const int tid = threadIdx.x;
  const int wg  = blockIdx.x;
  const int b   = wg >> 7;          // 128 tiles per image
  const int tx  = wg & 7;           // 8 tiles in x
  const int ty  = (wg >> 3) & 15;   // 16 tiles in y
  const int xb  = tx * TW;
  const int yb  = ty * THT;

  const float m00 = M[0], m01 = M[1], m02 = M[2];
  const float m10 = M[3], m11 = M[4], m12 = M[5];
  const float m20 = M[6], m21 = M[7], m22 = M[8];
  // adjugate of M: Minv = adj/det and det cancels in u = s0/s2, v = s1/s2.
  const float a00 = m11*m22 - m12*m21, a01 = m02*m21 - m01*m22, a02 = m01*m12 - m02*m11;
  const float a10 = m12*m20 - m10*m22, a11 = m00*m22 - m02*m20, a12 = m02*m10 - m00*m12;
  const float a20 = m10*m21 - m11*m20, a21 = m01*m20 - m00*m21, a22 = m00*m11 - m01*m10;

  // Source bbox for this tile: projective map is fractional-linear, so u,v
  // extrema over the tile rectangle occur at its 4 corners.
  float umin = 1e30f, umax = -1e30f, vmin = 1e30f, vmax = -1e30f;
#pragma unroll
  for (int cy = 0; cy < 2; ++cy)
#pragma unroll
    for (int cx = 0; cx < 2; ++cx) {
      float x = (float)(xb + cx * (TW - 1));
      float y = (float)(yb + cy * (THT - 1));
      float s2 = a20 * x + a21 * y + a22;
      float iu = (a00 * x + a01 * y + a02) / s2;
      float iv = (a10 * x + a11 * y + a12) / s2;
      umin = fminf(umin, iu); umax = fmaxf(umax, iu);
      vmin = fminf(vmin, iv); vmax = fmaxf(vmax, iv);
    }
  // bound before int conversion (NaN collapses via fmin/fmax -> fallback-safe)
  umin = fminf(fmaxf(umin, -1e6f), 1e6f);  umax = fminf(fmaxf(umax, -1e6f), 1e6f);
  vmin = fminf(fmaxf(vmin, -1e6f), 1e6f);  vmax = fminf(fmaxf(vmax, -1e6f), 1e6f);
  int bx0 = max((int)floorf(umin), 0);
  int bx1 = min((int)floorf(umax) + 1, WI - 1);
  int by0 = max((int)floorf(vmin), 0);
  int by1 = min((int)floorf(vmax) + 1, HI - 1);
  if (bx1 < bx0) { bx0 = 0; bx1 = 0; }
  if (by1 < by0) { by0 = 0; by1 = 0; }
  const int bw    = bx1 - bx0 + 1;
  const int bh    = by1 - by0 + 1;
  const int pitch = (bw * CI + 3) & ~3;                 // floats, mult of 4
  const bool useLds = (pitch * bh <= LDSF);             // uniform per block

  const float* gimg   = imgs + (size_t)b * (HI * WI * CI);
  float*       dstImg = out  + (size_t)b * (HI * WI * CI);

  if (useLds) {
    // ---- async-stage bbox rows into LDS (b128 per lane) ----
    const int  nChunks = (pitch * bh) >> 2;
    const long maxOfs  = (long)HI * WI * CI - 4;        // clamp tail over-read
    for (int c = tid; c < nChunks; c += 256) {
      int  f   = c << 2;
      int  r   = f / pitch;
      int  col = f - r * pitch;
      long g   = (long)(by0 + r) * (WI * CI) + (long)bx0 * CI + col;
      if (g > maxOfs) g = maxOfs;                       // pad slots never read
      g2l_b128(gimg + g, &lds[f]);
    }
    async_wait_all();
    __syncthreads();

#pragma unroll
    for (int k = 0; k < (TW * THT) / 256; ++k) {
      int p  = tid + k * 256;
      int lx = p & (TW - 1);
      int ly = p >> 6;                                  // log2(TW)
      float x = (float)(xb + lx);
      float y = (float)(yb + ly);
      float inv = 1.0f / (a20 * x + a21 * y + a22);
      float u = (a00 * x + a01 * y + a02) * inv;
      float v = (a10 * x + a11 * y + a12) * inv;
      bilinear3(lds, pitch, bx0, by0, bx0, bx1, by0, by1, u, v,
                dstImg + ((size_t)(yb + ly) * WI + (xb + lx)) * CI);
    }
  } else {
    // ---- fallback: direct global gather (pathological M only) ----
#pragma unroll
    for (int k = 0; k < (TW * THT) / 256; ++k) {
      int p  = tid + k * 256;
      int lx = p & (TW - 1);
      int ly = p >> 6;
      float x = (float)(xb + lx);
      float y = (float)(yb + ly);
      float inv = 1.0f / (a20 * x + a21 * y + a22);
      float u = (a00 * x + a01 * y + a02) * inv;
      float v = (a10 * x + a11 * y + a12) * inv;
      bilinear3(gimg, WI * CI, 0, 0, 0, WI - 1, 0, HI - 1, u, v,
                dstImg + ((size_t)(yb + ly) * WI + (xb + lx)) * CI);
    }
  }
}

extern "C" void kernel_launch(void* const* d_in, const int* in_sizes, int n_in,
                              void* d_out, int out_size, void* d_ws, size_t ws_size,
                              hipStream_t stream) {
  (void)in_sizes; (void)n_in; (void)out_size; (void)d_ws; (void)ws_size;
  const float* imgs = (const float*)d_in[0];   // (64,512,512,3) f32
  const float* M    = (const float*)d_in[1];   // (3,3) f32
  float*       out  = (float*)d_out;           // (64,512,512,3) f32
  const int nBlocks = BI * (WI / TW) * (HI / THT);   // 64 * 8 * 16 = 8192
  warp_persp_tiled<<<nBlocks, 256, 0, stream>>>(imgs, M, out);
}